// Criterion_15539191677204
// MI455X (gfx1250) — compile-verified
//
#include <hip/hip_runtime.h>

typedef float v2f __attribute__((ext_vector_type(2)));
typedef float v8f __attribute__((ext_vector_type(8)));

#define NB     32          // batches
#define HH     512
#define WW     512
#define HWSZ   (HH * WW)   // 262144 elements per image
#define CHUNKS 32          // blocks per batch image
#define TPB    256
#define QUADS_PER_BLOCK (HWSZ / CHUNKS / 4)   // 2048 float4 quads per block
#define ITERS  (QUADS_PER_BLOCK / TPB)        // 8 quads per thread

// d_ws layout (floats):
//   [0    .. 1023]  per-block sum of (o-d)^2
//   [1024 .. 2047]  per-block sum of (o-d)
//   [2048 .. 5119]  per-block 3 box partial sums
__global__ __launch_bounds__(TPB) void criterion_partials(
    const float* __restrict__ outp, const float* __restrict__ dmap,
    const int* __restrict__ bboxes, float* __restrict__ ws)
{
    const int blk   = blockIdx.x;
    const int b     = blk / CHUNKS;
    const int chunk = blk % CHUNKS;
    const int tid   = threadIdx.x;

    const float4* o4 = reinterpret_cast<const float4*>(outp + (size_t)b * HWSZ);
    const float4* d4 = reinterpret_cast<const float4*>(dmap + (size_t)b * HWSZ);

    // Load & clamp this batch's 3 boxes (x1,y1,x2,y2), per reference semantics.
    int bx1[3], by1[3], bx2[3], by2[3];
    const int* bb = bboxes + b * 12;
#pragma unroll
    for (int k = 0; k < 3; ++k) {
        int x1 = bb[k * 4 + 0], y1 = bb[k * 4 + 1];
        int x2 = bb[k * 4 + 2], y2 = bb[k * 4 + 3];
        x1 = min(max(x1, 0), WW); y1 = min(max(y1, 0), HH);
        x2 = min(max(x2, 0), WW); y2 = min(max(y2, 0), HH);
        x2 = max(x2, x1); y2 = max(y2, y1);
        bx1[k] = x1; by1[k] = y1; bx2[k] = x2; by2[k] = y2;
    }

    // Per-lane x is loop-invariant: each iteration advances 1024 elements (== 0 mod 512).
    // Precompute x-membership for the 4 elements of this lane's quad, per box.
    const int x0 = (4 * tid) & (WW - 1);
    bool xin[3][4];
#pragma unroll
    for (int k = 0; k < 3; ++k)
#pragma unroll
        for (int j = 0; j < 4; ++j)
            xin[k][j] = (x0 + j >= bx1[k]) & (x0 + j < bx2[k]);

    // WMMA accumulators: D = A(16x4 f32) * ones(4x16) + C.
    // Every A element lands in 16 accumulator slots -> final scale 1/16.
    v8f acc_df = {0.f, 0.f, 0.f, 0.f, 0.f, 0.f, 0.f, 0.f};
    v8f acc_sq = {0.f, 0.f, 0.f, 0.f, 0.f, 0.f, 0.f, 0.f};
    v2f ones;  ones.x = 1.0f; ones.y = 1.0f;
    float box0 = 0.f, box1 = 0.f, box2 = 0.f;

    const int quad_base = chunk * QUADS_PER_BLOCK;
#pragma unroll
    for (int i = 0; i < ITERS; ++i) {
        const int q = quad_base + i * TPB + tid;    // float4 index within image
        const float4 o = o4[q];
        const float4 d = d4[q];
        const float df0 = o.x - d.x, df1 = o.y - d.y;
        const float df2 = o.z - d.z, df3 = o.w - d.w;

        v2f a;
        a.x = df0;       a.y = df1;
        acc_df = __builtin_amdgcn_wmma_f32_16x16x4_f32(false, a, false, ones,
                                                       (short)0, acc_df, false, false);
        a.x = df2;       a.y = df3;
        acc_df = __builtin_amdgcn_wmma_f32_16x16x4_f32(false, a, false, ones,
                                                       (short)0, acc_df, false, false);
        a.x = df0 * df0; a.y = df1 * df1;
        acc_sq = __builtin_amdgcn_wmma_f32_16x16x4_f32(false, a, false, ones,
                                                       (short)0, acc_sq, false, false);
        a.x = df2 * df2; a.y = df3 * df3;
        acc_sq = __builtin_amdgcn_wmma_f32_16x16x4_f32(false, a, false, ones,
                                                       (short)0, acc_sq, false, false);

        // Box membership: y varies per iteration (and between the block's two 128-thread halves); x precomputed.
        const int e = q * 4;
        const int y = (e >> 9) & (HH - 1);
        const bool yin0 = (y >= by1[0]) & (y < by2[0]);
        const bool yin1 = (y >= by1[1]) & (y < by2[1]);
        const bool yin2 = (y >= by1[2]) & (y < by2[2]);
        box0 += (yin0 & xin[0][0]) ? o.x : 0.f;
        box0 += (yin0 & xin[0][1]) ? o.y : 0.f;
        box0 += (yin0 & xin[0][2]) ? o.z : 0.f;
        box0 += (yin0 & xin[0][3]) ? o.w : 0.f;
        box1 += (yin1 & xin[1][0]) ? o.x : 0.f;
        box1 += (yin1 & xin[1][1]) ? o.y : 0.f;
        box1 += (yin1 & xin[1][2]) ? o.z : 0.f;
        box1 += (yin1 & xin[1][3]) ? o.w : 0.f;
        box2 += (yin2 & xin[2][0]) ? o.x : 0.f;
        box2 += (yin2 & xin[2][1]) ? o.y : 0.f;
        box2 += (yin2 & xin[2][2]) ? o.z : 0.f;
        box2 += (yin2 & xin[2][3]) ? o.w : 0.f;
    }

    // Collapse the 8 accumulator VGPRs per lane.
    float l_df = 0.f, l_sq = 0.f;
#pragma unroll
    for (int r = 0; r < 8; ++r) { l_df += acc_df[r]; l_sq += acc_sq[r]; }

    // wave32 reduction
#pragma unroll
    for (int off = 16; off > 0; off >>= 1) {
        l_df += __shfl_down(l_df, off, 32);
        l_sq += __shfl_down(l_sq, off, 32);
        box0 += __shfl_down(box0, off, 32);
        box1 += __shfl_down(box1, off, 32);
        box2 += __shfl_down(box2, off, 32);
    }

    __shared__ float s_df[8], s_sq[8], s_bx[8][3];
    const int wave = tid >> 5, lane = tid & 31;
    if (lane == 0) {
        s_df[wave] = l_df; s_sq[wave] = l_sq;
        s_bx[wave][0] = box0; s_bx[wave][1] = box1; s_bx[wave][2] = box2;
    }
    __syncthreads();
    if (tid == 0) {
        float t_df = 0.f, t_sq = 0.f, t0 = 0.f, t1 = 0.f, t2 = 0.f;
#pragma unroll
        for (int w = 0; w < 8; ++w) {
            t_df += s_df[w]; t_sq += s_sq[w];
            t0 += s_bx[w][0]; t1 += s_bx[w][1]; t2 += s_bx[w][2];
        }
        ws[blk]                = t_sq * (1.0f / 16.0f);   // undo 16x WMMA replication
        ws[1024 + blk]         = t_df * (1.0f / 16.0f);
        ws[2048 + blk * 3 + 0] = t0;
        ws[2048 + blk * 3 + 1] = t1;
        ws[2048 + blk * 3 + 2] = t2;
    }
}

__global__ __launch_bounds__(1024) void criterion_finalize(
    const float* __restrict__ ws, const int* __restrict__ nobj,
    float* __restrict__ out)
{
    __shared__ float red[32];
    const int t = threadIdx.x;          // 0..1023
    const int wave = t >> 5, lane = t & 31;

    // ---- dmap loss: sum all 1024 sq partials ----
    float v = ws[t];
#pragma unroll
    for (int off = 16; off > 0; off >>= 1) v += __shfl_down(v, off, 32);
    if (lane == 0) red[wave] = v;
    __syncthreads();
    if (wave == 0) {
        float x = red[lane];
#pragma unroll
        for (int off = 16; off > 0; off >>= 1) x += __shfl_down(x, off, 32);
        if (lane == 0) out[0] = x / (float)(*nobj);
    }
    __syncthreads();

    // ---- count loss: wave w owns batch w (its 32 chunk partials are contiguous) ----
    float d = ws[1024 + t];
#pragma unroll
    for (int off = 16; off > 0; off >>= 1) d += __shfl_down(d, off, 32);
    if (lane == 0) red[wave] = d * d;   // (sum_b(o) - sum_b(d))^2
    __syncthreads();
    if (wave == 0) {
        float x = red[lane];
#pragma unroll
        for (int off = 16; off > 0; off >>= 1) x += __shfl_down(x, off, 32);
        if (lane == 0) out[1] = x * (1.0f / (float)NB);
    }
    __syncthreads();

    // ---- min-count loss: 96 (batch,box) pairs, each sums 32 chunk partials ----
    float mc = 0.f;
    if (t < NB * 3) {
        const int b = t / 3, k = t % 3;
        float s = 0.f;
        for (int c = 0; c < CHUNKS; ++c)
            s += ws[2048 + (b * CHUNKS + c) * 3 + k];
        mc = fmaxf(1.0f - s, 0.0f);
    }
#pragma unroll
    for (int off = 16; off > 0; off >>= 1) mc += __shfl_down(mc, off, 32);
    if (lane == 0) red[wave] = mc;
    __syncthreads();
    if (wave == 0) {
        float x = red[lane];
#pragma unroll
        for (int off = 16; off > 0; off >>= 1) x += __shfl_down(x, off, 32);
        if (lane == 0) out[2] = x;
    }
}

extern "C" void kernel_launch(void* const* d_in, const int* in_sizes, int n_in,
                              void* d_out, int out_size, void* d_ws, size_t ws_size,
                              hipStream_t stream) {
    const float* output = (const float*)d_in[0];
    const float* dmap   = (const float*)d_in[1];
    const int*   bboxes = (const int*)d_in[2];   // (B,3,4)
    const int*   nobj   = (const int*)d_in[3];   // scalar num_objects
    float* ws  = (float*)d_ws;
    float* out = (float*)d_out;

    criterion_partials<<<NB * CHUNKS, TPB, 0, stream>>>(output, dmap, bboxes, ws);
    criterion_finalize<<<1, 1024, 0, stream>>>(ws, nobj, out);
}